// GCNLayer_72765335929214
// MI455X (gfx1250) — compile-verified
//
#include <hip/hip_runtime.h>
#include <hip/hip_bf16.h>

typedef float v2f __attribute__((ext_vector_type(2)));
typedef float v8f __attribute__((ext_vector_type(8)));

#define D 128           // D_IN == D_OUT == 128
#define N_NODES_C 50000

// ---------------------------------------------------------------------------
// Kernel 0: zero the aggregation workspace h [N_NODES x 128] (f32) in d_ws.
// ---------------------------------------------------------------------------
__global__ void gcn_zero_kernel(float4* __restrict__ p, int n4) {
    int i = blockIdx.x * blockDim.x + threadIdx.x;
    if (i < n4) p[i] = make_float4(0.f, 0.f, 0.f, 0.f);
}

// ---------------------------------------------------------------------------
// Kernel 1: edge-parallel gather + scatter-add.
// One wave32 per edge: lane L loads feature[src][4L..4L+3] as a b128 and
// issues 4 non-returning global_atomic_add_f32 into h[dst]. Both feature
// (25.6MB) and h (25.6MB) are L2-resident on MI455X (192MB L2), so the
// atomics resolve at L2 bandwidth.
// ---------------------------------------------------------------------------
__global__ void gcn_scatter_kernel(const float* __restrict__ feature,
                                   const int*   __restrict__ edge_src,
                                   const int*   __restrict__ edge_dst,
                                   float*       __restrict__ h,
                                   int n_edges) {
    const int wavesPerBlock = blockDim.x >> 5;
    const int e    = blockIdx.x * wavesPerBlock + (threadIdx.x >> 5);
    const int lane = threadIdx.x & 31;
    if (e >= n_edges) return;

    const int s = edge_src[e];
    const int d = edge_dst[e];

    const float4 v = ((const float4*)(feature + (size_t)s * D))[lane];
    float* hp = h + (size_t)d * D + lane * 4;
    atomicAdd(hp + 0, v.x);
    atomicAdd(hp + 1, v.y);
    atomicAdd(hp + 2, v.z);
    atomicAdd(hp + 3, v.w);
}

// ---------------------------------------------------------------------------
// Kernel 2: out = relu(h @ W + b) with V_WMMA_F32_16X16X4_F32.
// One wave per 16-row M-tile; it accumulates all 8 N-tiles (8 x v8f = 64
// VGPRs) while streaming A exactly once. W (64KB) stays cache resident.
//
// f32 WMMA VGPR layouts (ISA 7.12.2):
//   A 16x4 : lanes 0-15 -> M=lane,   K = {0,1}; lanes 16-31 -> M=lane-16, K={2,3}
//   B 4x16 : VGPR0 -> K=0 (lanes 0-15) / K=2 (lanes 16-31); VGPR1 -> K=1 / K=3
//   C 16x16: VGPR r -> row r (lanes 0-15) and row r+8 (lanes 16-31), col=lane&15
// ---------------------------------------------------------------------------
__global__ void gcn_gemm_kernel(const float* __restrict__ h,
                                const float* __restrict__ W,
                                const float* __restrict__ bias,
                                float*       __restrict__ out,
                                int n_mtiles) {
    const int wavesPerBlock = blockDim.x >> 5;
    const int mtile = blockIdx.x * wavesPerBlock + (threadIdx.x >> 5);
    const int lane  = threadIdx.x & 31;
    if (mtile >= n_mtiles) return;   // wave-uniform: EXEC stays all-1s for WMMA

    const int hi    = lane >> 4;        // 0 for lanes 0-15, 1 for lanes 16-31
    const int khalf = hi << 1;          // 0 or 2
    const int ncol  = lane & 15;
    const int arow  = (mtile << 4) + (lane & 15);

    v8f acc[8];
#pragma unroll
    for (int i = 0; i < 8; ++i) acc[i] = (v8f)(0.0f);

    const float* arowp = h + (size_t)arow * D;

    for (int k = 0; k < D; k += 4) {
        // A tile: contiguous pair, 8-byte aligned
        v2f a = *(const v2f*)(arowp + k + khalf);
#pragma unroll
        for (int nt = 0; nt < 8; ++nt) {
            const float* bp = W + (size_t)(k + khalf) * D + nt * 16 + ncol;
            v2f b;
            b.x = bp[0];
            b.y = bp[D];
            acc[nt] = __builtin_amdgcn_wmma_f32_16x16x4_f32(
                /*neg_a=*/false, a, /*neg_b=*/false, b,
                /*c_mod=*/(short)0, acc[nt],
                /*reuse_a=*/false, /*reuse_b=*/false);
        }
    }

    // Epilogue: bias + ReLU, store per ISA C/D layout
#pragma unroll
    for (int nt = 0; nt < 8; ++nt) {
        const int n = nt * 16 + ncol;
        const float bv = bias[n];
#pragma unroll
        for (int r = 0; r < 8; ++r) {
            const int m = (mtile << 4) + (hi << 3) + r;
            float v = acc[nt][r] + bv;
            out[(size_t)m * D + n] = v > 0.f ? v : 0.f;
        }
    }
}

// ---------------------------------------------------------------------------
// Launcher
// inputs: 0=feature[N,128] f32, 1=edge_src[E] i32, 2=edge_dst[E] i32,
//         3=W[128,128] f32, 4=b[128] f32.  d_out: [N,128] f32.
// d_ws: h accumulator, N*128*4 = 25.6 MB.
// ---------------------------------------------------------------------------
extern "C" void kernel_launch(void* const* d_in, const int* in_sizes, int n_in,
                              void* d_out, int out_size, void* d_ws, size_t ws_size,
                              hipStream_t stream) {
    const float* feature  = (const float*)d_in[0];
    const int*   edge_src = (const int*)d_in[1];
    const int*   edge_dst = (const int*)d_in[2];
    const float* W        = (const float*)d_in[3];
    const float* bias     = (const float*)d_in[4];
    float*       out      = (float*)d_out;

    const int n_nodes = in_sizes[0] / D;     // 50000
    const int n_edges = in_sizes[1];         // 800000

    float* h = (float*)d_ws;                 // [n_nodes, 128] f32 accumulator

    // 1) zero h
    {
        const int n4 = n_nodes * D / 4;
        const int threads = 256;
        const int blocks = (n4 + threads - 1) / threads;
        gcn_zero_kernel<<<blocks, threads, 0, stream>>>((float4*)h, n4);
    }

    // 2) scatter-add: one wave per edge, 8 waves (256 threads) per block
    {
        const int threads = 256;
        const int wavesPerBlock = threads / 32;
        const int blocks = (n_edges + wavesPerBlock - 1) / wavesPerBlock;
        gcn_scatter_kernel<<<blocks, threads, 0, stream>>>(feature, edge_src,
                                                           edge_dst, h, n_edges);
    }

    // 3) WMMA GEMM + bias + relu: one wave per 16-row tile
    {
        const int n_mtiles = n_nodes / 16;   // 3125
        const int threads = 256;
        const int wavesPerBlock = threads / 32;
        const int blocks = (n_mtiles + wavesPerBlock - 1) / wavesPerBlock;
        gcn_gemm_kernel<<<blocks, threads, 0, stream>>>(h, W, bias, out, n_mtiles);
    }
}